// ValueLayerSlow_1245540516057
// MI455X (gfx1250) — compile-verified
//
#include <hip/hip_runtime.h>

// ---------------------------------------------------------------------------
// Pauli expectation values on 10 qubits, real states.
//   out[n, o<10]   = sum_d (-1)^{bit_{9-o}(d)} x[n,d]^2          (Z ops)
//   out[n, 10..12] = sum_d x[n,d] * x[n, d ^ (512>>(o-10))]      (X ops)
//   out[n, 13..15] = 0                                           (Y ops, real psi)
//
// Memory-bound: 16 MiB in / 23.3 TB/s ~ 0.7 us. WMMA f32 16x16x4 is used as a
// 16-op x 16-state weighted-reduction engine: A holds the +-1 sign patterns
// (Z) / one-hot rows (X), B holds per-state data (x^2 or x*x_partner).
// Stage 1 uses CDNA5 async global->LDS (ASYNCcnt) to avoid the VGPR round
// trip on the bandwidth-critical path.
// ---------------------------------------------------------------------------

typedef float v2f __attribute__((ext_vector_type(2)));
typedef float v8f __attribute__((ext_vector_type(8)));

#define DIM 1024
#define TILE_STATES 16
#define N_OPS 16

__launch_bounds__(128, 1)
__global__ void pauli_expect_kernel(const float* __restrict__ x,
                                    float* __restrict__ out) {
  // 16 states x 1024 cols, XOR-swizzled by (row<<2) in word-address bits 2..5
  // so that 16 lanes streaming 16 different rows hit 16 distinct LDS banks.
  __shared__ __align__(16) float tile[TILE_STATES * DIM];  // 64 KB

  const int tid = threadIdx.x;
  const int n0  = blockIdx.x * TILE_STATES;

  // ---- Stage 1: async global -> LDS, coalesced + bank-swizzled ----
  {
    const float* xbase = x + (size_t)n0 * DIM;
    // 16*1024/4 = 4096 16B-transfers / 128 threads = 32 per thread.
#pragma unroll 4
    for (int i = 0; i < 32; ++i) {
      int g  = i * 128 + tid;
      int r  = g >> 8;    // row 0..15 (constant within an iteration)
      int c4 = g & 255;   // float4 column
      // swizzle touches word-addr bits 2..5 only -> 16B stays contiguous/aligned
      unsigned lds_addr =
          (unsigned)(uintptr_t)&tile[r * DIM + ((c4 * 4) ^ (r << 2))];
      int goff = g * 16;  // byte offset from xbase, per lane
      asm volatile("global_load_async_to_lds_b128 %0, %1, %2"
                   :
                   : "v"(lds_addr), "v"(goff), "s"(xbase)
                   : "memory");
    }
    asm volatile("s_wait_asynccnt 0" ::: "memory");
  }
  __syncthreads();

  // ---- Stage 2: WMMA reduction chains ----
  const int l    = tid & 31;   // lane
  const int w    = tid >> 5;   // wave 0: Z ops; waves 1..3: X masks 512/256/128
  const int half = l >> 4;
  const int r    = l & 15;     // doubles as: B column N (state), A row M (op)
  const int swz  = r << 2;
  const int off  = 2 * half;   // K slots 2*half, 2*half+1 per the 16x4 layout
  const float* row = &tile[r * DIM];

  v8f acc = {};  // D/C: 16x16 f32, lane l holds N=l%16, M = vgpr + 8*(l/16)

  if (w == 0) {
    // Z segment: B[k][n] = x[n,k]^2 ; A[o][k] = (-1)^{bit_{9-o}(k)} for o<10 else 0.
    const int   sh = (r < 10) ? (9 - r) : 0;
    const float zm = (r < 10) ? 1.0f : 0.0f;
#pragma unroll 8
    for (int kb = 0; kb < DIM; kb += 4) {
      int   k0 = kb + off;
      int   i0 = k0 ^ swz;
      float v0 = row[i0];
      float v1 = row[i0 + 1];
      v2f a, b;
      b[0] = v0 * v0;
      b[1] = v1 * v1;
      a[0] = (( k0      >> sh) & 1) ? -zm : zm;
      a[1] = (((k0 + 1) >> sh) & 1) ? -zm : zm;
      acc = __builtin_amdgcn_wmma_f32_16x16x4_f32(false, a, false, b,
                                                  (short)0, acc, false, false);
    }
  } else {
    // X segment: B[k][n] = x[n,k]*x[n,k^mask] ; A one-hot on row o = 9+w.
    const int   mask = 512 >> (w - 1);         // 512, 256, 128 (bits >= 7)
    const float av   = (r == 9 + w) ? 1.0f : 0.0f;
    v2f a;
    a[0] = av;
    a[1] = av;
#pragma unroll 8
    for (int kb = 0; kb < DIM; kb += 4) {
      int   k0 = kb + off;
      int   i0 = k0 ^ swz;       // mask bits (>=7) and swz bits (2..5) commute
      int   p0 = i0 ^ mask;
      float v0 = row[i0];
      float v1 = row[i0 + 1];
      float u0 = row[p0];
      float u1 = row[p0 + 1];
      v2f b;
      b[0] = v0 * u0;
      b[1] = v1 * u1;
      acc = __builtin_amdgcn_wmma_f32_16x16x4_f32(false, a, false, b,
                                                  (short)0, acc, false, false);
    }
  }

  // ---- Epilogue: each wave stores its disjoint op rows of D ----
  // D: lane l -> state n = l%16; vgpr j -> op o = j + 8*(l/16).
  float* op = out + (size_t)(n0 + r) * N_OPS;
#pragma unroll
  for (int j = 0; j < 8; ++j) {
    int  o    = j + 8 * half;
    bool mine = (w == 0) ? (o < 10 || o >= 13)   // Z rows + zero Y rows
                         : (o == 9 + w);         // this wave's X row
    if (mine) op[o] = acc[j];
  }
}

extern "C" void kernel_launch(void* const* d_in, const int* in_sizes, int n_in,
                              void* d_out, int out_size, void* d_ws, size_t ws_size,
                              hipStream_t stream) {
  const float* x = (const float*)d_in[0];
  float* out     = (float*)d_out;
  const int n_states = in_sizes[0] / DIM;          // 8*512 = 4096
  dim3 grid(n_states / TILE_STATES);               // 256 blocks
  dim3 block(128);                                 // 4 wave32
  pauli_expect_kernel<<<grid, block, 0, stream>>>(x, out);
}